// ArticulationTorch_57913339019686
// MI455X (gfx1250) — compile-verified
//
#include <hip/hip_runtime.h>

#define B_  256
#define J_  140
#define VN_ 14000
#define TILES_ ((VN_ + 15) / 16)   // 875, exact

typedef float v2f __attribute__((ext_vector_type(2)));
typedef float v8f __attribute__((ext_vector_type(8)));

// ---------------------------------------------------------------------------
// Kernel 1: Rodrigues + forward-kinematics chain + A matrices + J_final.
// One block per batch. S/G live in LDS; wave 0 walks the serial parent chain.
// ---------------------------------------------------------------------------
__global__ void __launch_bounds__(64) fk_kernel(
    const float* __restrict__ rotvec,   // (B,J,3)
    const float* __restrict__ ptrans,   // (B,J,3)
    const float* __restrict__ tpose,    // (J,3)
    const int*   __restrict__ parents,  // (J,)
    float* __restrict__ A_ws,           // (B,J,12) scratch
    float* __restrict__ Jf)             // (B,J,3)
{
    __shared__ float S[J_][12];
    __shared__ float G[J_][12];
    const int b   = blockIdx.x;
    const int tid = threadIdx.x;

    // Phase 1: per-joint local transforms (Rodrigues), parallel over joints.
    for (int j = tid; j < J_; j += 64) {
        const int base = (b * J_ + j) * 3;
        float rx = rotvec[base + 0], ry = rotvec[base + 1], rz = rotvec[base + 2];
        float theta = sqrtf(rx * rx + ry * ry + rz * rz) + 1e-8f;
        float inv = 1.0f / theta;
        float x = rx * inv, y = ry * inv, z = rz * inv;
        float c = cosf(theta), s = sinf(theta), t1 = 1.0f - c;
        S[j][0] = c + t1 * x * x;   S[j][1] = t1 * x * y - s * z; S[j][2]  = t1 * x * z + s * y;
        S[j][4] = t1 * y * x + s*z; S[j][5] = c + t1 * y * y;     S[j][6]  = t1 * y * z - s * x;
        S[j][8] = t1 * z * x - s*y; S[j][9] = t1 * z * y + s * x; S[j][10] = c + t1 * z * z;
        S[j][3]  = ptrans[base + 0];
        S[j][7]  = ptrans[base + 1];
        S[j][11] = ptrans[base + 2];
    }
    __syncthreads();

    // Phase 2: serial chain G[j] = G[parent[j]] @ S[j], wave 0, 12 lanes active.
    if (tid < 32) {
        const int lane = tid;
        if (lane < 12) G[0][lane] = S[0][lane];
        for (int j = 1; j < J_; ++j) {
            int p = parents[j];
            if (lane < 12) {
                int m = lane >> 2, n = lane & 3;
                float acc = G[p][m * 4 + 0] * S[j][0 * 4 + n]
                          + G[p][m * 4 + 1] * S[j][1 * 4 + n]
                          + G[p][m * 4 + 2] * S[j][2 * 4 + n];
                if (n == 3) acc += G[p][m * 4 + 3];
                G[j][lane] = acc;   // same-wave LDS is in-order (DScnt)
            }
        }
    }
    __syncthreads();

    // Phase 3: A = [R | t - R*tpose], J_final = t.
    for (int j = tid; j < J_; j += 64) {
        float tx = tpose[j * 3 + 0], ty = tpose[j * 3 + 1], tz = tpose[j * 3 + 2];
        for (int m = 0; m < 3; ++m) {
            float r0 = G[j][m * 4 + 0], r1 = G[j][m * 4 + 1];
            float r2 = G[j][m * 4 + 2], g3 = G[j][m * 4 + 3];
            float* Ap = A_ws + (size_t)(b * J_ + j) * 12 + m * 4;
            Ap[0] = r0; Ap[1] = r1; Ap[2] = r2;
            Ap[3] = g3 - (r0 * tx + r1 * ty + r2 * tz);
            Jf[(b * J_ + j) * 3 + m] = g3;
        }
    }
}

// ---------------------------------------------------------------------------
// Kernel 2: fp32 WMMA skinning.  Per wave: one 16-vertex tile of one batch.
// T-tile = W(16x140) @ A(140x16pad) via 35x v_wmma_f32_16x16x4_f32, then the
// T . [v,1] contraction is fused in-register/LDS -> V_final directly.
// ---------------------------------------------------------------------------
__global__ void __launch_bounds__(128) skin_wmma_kernel(
    const float* __restrict__ weights,  // (VN,J)
    const float* __restrict__ vtempl,   // (VN,3)
    const float* __restrict__ A_ws,     // (B,J,12)
    float* __restrict__ Vf)             // (B,VN,3)
{
    __shared__ __align__(16) float Apad[16][144];  // [col n][joint k], zero-padded
    __shared__ __align__(16) float Tt[4][16][16];  // per-wave T tile

    const int b   = blockIdx.y;
    const int tid = threadIdx.x;

    // Stage per-batch A into LDS, transposed + padded (cols 12..15, k 140..143 = 0).
    const float* Ab = A_ws + (size_t)b * J_ * 12;
    for (int i = tid; i < 16 * 144; i += 128) {
        int n = i / 144, k = i % 144;
        Apad[n][k] = (n < 12 && k < J_) ? Ab[k * 12 + n] : 0.0f;
    }
    __syncthreads();

    const int wave = tid >> 5;
    const int lane = tid & 31;
    const int tile = blockIdx.x * 4 + wave;
    if (tile >= TILES_) return;              // wave-uniform: EXEC stays all-ones

    const int vbase = tile * 16;
    const int half  = lane >> 4;             // 0: K={k0,k0+1}, 1: K={k0+2,k0+3}
    const int mrow  = lane & 15;             // A: M row / B: N col
    const int koff  = half * 2;

    const float* wrow = weights + (size_t)(vbase + mrow) * J_ + koff;  // 8B aligned
    const float* brow = &Apad[mrow][koff];                             // 8B aligned

    v8f acc = {};
    #pragma unroll
    for (int k0 = 0; k0 < J_; k0 += 4) {
        v2f a  = *reinterpret_cast<const v2f*>(wrow + k0);
        v2f bb = *reinterpret_cast<const v2f*>(brow + k0);
        acc = __builtin_amdgcn_wmma_f32_16x16x4_f32(
            /*neg_a=*/false, a, /*neg_b=*/false, bb,
            /*c_mod=*/(short)0, acc, /*reuse_a=*/false, /*reuse_b=*/false);
    }

    // Accumulator layout: lane<16 -> rows 0..7, lane>=16 -> rows 8..15; col = mrow.
    #pragma unroll
    for (int r = 0; r < 8; ++r)
        Tt[wave][r + (half << 3)][mrow] = acc[r];
    // Intra-wave producer/consumer through LDS: DS ops are in-order, no barrier.

    if (lane < 16) {
        int v = vbase + lane;
        float vx = vtempl[v * 3 + 0], vy = vtempl[v * 3 + 1], vz = vtempl[v * 3 + 2];
        #pragma unroll
        for (int m = 0; m < 3; ++m) {
            float val = Tt[wave][lane][4 * m + 0] * vx
                      + Tt[wave][lane][4 * m + 1] * vy
                      + Tt[wave][lane][4 * m + 2] * vz
                      + Tt[wave][lane][4 * m + 3];
            Vf[((size_t)b * VN_ + v) * 3 + m] = val;
        }
    }
}

// ---------------------------------------------------------------------------
// Kernel 3: keypoints. kp_is_joint = (k % 2 == 0) by construction.
// ---------------------------------------------------------------------------
__global__ void __launch_bounds__(128) keypoint_kernel(
    const float* __restrict__ Vf, const float* __restrict__ Jf,
    const int* __restrict__ kp_v_ids, const int* __restrict__ kp_j_ids,
    float* __restrict__ kp)
{
    int idx = blockIdx.x * 128 + threadIdx.x;
    if (idx >= B_ * 22) return;
    int b = idx / 22, k = idx % 22;
    float o0, o1, o2;
    if ((k & 1) == 0) {
        int j0 = kp_j_ids[k * 2 + 0], j1 = kp_j_ids[k * 2 + 1];
        const float* p0 = Jf + (b * J_ + j0) * 3;
        const float* p1 = Jf + (b * J_ + j1) * 3;
        o0 = 0.5f * (p0[0] + p1[0]);
        o1 = 0.5f * (p0[1] + p1[1]);
        o2 = 0.5f * (p0[2] + p1[2]);
    } else {
        o0 = o1 = o2 = 0.0f;
        for (int q = 0; q < 4; ++q) {
            int v = kp_v_ids[k * 4 + q];
            const float* p = Vf + ((size_t)b * VN_ + v) * 3;
            o0 += p[0]; o1 += p[1]; o2 += p[2];
        }
        o0 *= 0.25f; o1 *= 0.25f; o2 *= 0.25f;
    }
    kp[idx * 3 + 0] = o0; kp[idx * 3 + 1] = o1; kp[idx * 3 + 2] = o2;
}

// ---------------------------------------------------------------------------
extern "C" void kernel_launch(void* const* d_in, const int* in_sizes, int n_in,
                              void* d_out, int out_size, void* d_ws, size_t ws_size,
                              hipStream_t stream) {
    const float* rotvec  = (const float*)d_in[0];   // pose_rot_vec
    const float* ptrans  = (const float*)d_in[1];   // pose_trans
    const float* vtempl  = (const float*)d_in[2];   // v_template
    const float* weights = (const float*)d_in[3];   // weights
    const float* tpose   = (const float*)d_in[4];   // tpose_joints
    const int*   parents = (const int*)d_in[5];     // parents
    const int*   kpv     = (const int*)d_in[6];     // kp_v_ids
    const int*   kpj     = (const int*)d_in[7];     // kp_j_ids
    // d_in[8] = kp_is_joint: arange(22)%2==0, used via parity directly.

    float* Vf = (float*)d_out;                       // (B,VN,3)
    float* Jf = Vf + (size_t)B_ * VN_ * 3;           // (B,J,3)
    float* kp = Jf + (size_t)B_ * J_ * 3;            // (B,22,3)
    float* A_ws = (float*)d_ws;                      // (B,J,12) = 1.72 MB

    fk_kernel<<<B_, 64, 0, stream>>>(rotvec, ptrans, tpose, parents, A_ws, Jf);

    dim3 g2((TILES_ + 3) / 4, B_);
    skin_wmma_kernel<<<g2, 128, 0, stream>>>(weights, vtempl, A_ws, Vf);

    keypoint_kernel<<<(B_ * 22 + 127) / 128, 128, 0, stream>>>(Vf, Jf, kpv, kpj, kp);
}